// GCN_82454782148692
// MI455X (gfx1250) — compile-verified
//
#include <hip/hip_runtime.h>
#include <hip/hip_bf16.h>

#define F 128          // F_IN == H == 128
#define EPS_F 1e-5f

typedef __attribute__((ext_vector_type(2))) float v2f;
typedef __attribute__((ext_vector_type(8))) float v8f;

// ---------------------------------------------------------------------------
// Degree / normalization
// ---------------------------------------------------------------------------
__global__ void k_deg_init(float* __restrict__ deg, int N) {
    int i = blockIdx.x * blockDim.x + threadIdx.x;
    if (i < N) deg[i] = 1.0f;                       // self-loop contribution
}

__global__ void k_deg_acc(const int* __restrict__ dst, float* __restrict__ deg, int E) {
    int i = blockIdx.x * blockDim.x + threadIdx.x;
    if (i < E) atomicAdd(&deg[dst[i]], 1.0f);
}

__global__ void k_dinv(float* __restrict__ deg, int N) {
    int i = blockIdx.x * blockDim.x + threadIdx.x;
    if (i < N) deg[i] = rsqrtf(deg[i]);             // deg >= 1 always (self-loops)
}

// ---------------------------------------------------------------------------
// WMMA fp32 GEMM:  Y[N x 128] = act(X)[N x 128] @ W[128 x 128]
// act = identity (FUSE_BN=false) or ReLU(BN(.)) fused into the LDS staging load.
// Block: 256 threads (8 waves), 64 rows per block. Wave w: row-strip w>>1,
// column-half w&1 (4 x 16x16 output tiles per wave). K=128 in steps of 4 via
// v_wmma_f32_16x16x4_f32 (full fp32 precision -> matches fp32 reference).
// Full-tile fast path: unguarded loads/stores, immediate-offset addressing.
// ---------------------------------------------------------------------------
#define LDSW 132       // 128 + 4 padding -> conflict-free b64 A-frag reads

template <bool FUSE_BN>
__device__ __forceinline__ float4 bn_relu4(float4 val, int c,
                                           const float* __restrict__ gg,
                                           const float* __restrict__ be,
                                           const float* __restrict__ mm,
                                           const float* __restrict__ vv) {
    if (FUSE_BN) {
        float4 G = *(const float4*)(gg + c);
        float4 B = *(const float4*)(be + c);
        float4 M = *(const float4*)(mm + c);
        float4 V = *(const float4*)(vv + c);
        val.x = fmaxf(0.f, (val.x - M.x) * (G.x * rsqrtf(V.x + EPS_F)) + B.x);
        val.y = fmaxf(0.f, (val.y - M.y) * (G.y * rsqrtf(V.y + EPS_F)) + B.y);
        val.z = fmaxf(0.f, (val.z - M.z) * (G.z * rsqrtf(V.z + EPS_F)) + B.z);
        val.w = fmaxf(0.f, (val.w - M.w) * (G.w * rsqrtf(V.w + EPS_F)) + B.w);
    }
    return val;
}

template <bool FUSE_BN>
__global__ void k_gemm(const float* __restrict__ X, const float* __restrict__ W,
                       float* __restrict__ Y, int N,
                       const float* __restrict__ gg, const float* __restrict__ be,
                       const float* __restrict__ mm, const float* __restrict__ vv) {
    __shared__ float lds[64 * LDSW];

    const int tid = threadIdx.x;
    const int rowbase = blockIdx.x * 64;
    const bool full = (rowbase + 64) <= N;           // block-uniform

    // Cooperative staging of the 64x128 X tile (coalesced b128 loads),
    // with optional fused BN+ReLU (exact: each element touched once).
    {
        int f4 = tid;                                // float4 index, 2048 total
        #pragma unroll
        for (int it = 0; it < 8; ++it, f4 += 256) {
            int r = (f4 * 4) >> 7;                   // row in tile
            int c = (f4 * 4) & 127;                  // col (channel)
            float4 val;
            if (full) {
                val = *(const float4*)(X + (size_t)(rowbase + r) * F + c);
            } else {
                val = make_float4(0.f, 0.f, 0.f, 0.f);
                if (rowbase + r < N)
                    val = *(const float4*)(X + (size_t)(rowbase + r) * F + c);
            }
            val = bn_relu4<FUSE_BN>(val, c, gg, be, mm, vv);
            *(float4*)&lds[r * LDSW + c] = val;      // 16B aligned
        }
    }
    __syncthreads();

    const int wave  = tid >> 5;
    const int lane  = tid & 31;
    const int strip = wave >> 1;                 // 0..3 : 16-row strip
    const int ch    = wave & 1;                  // 0..1 : column half
    const int half  = lane >> 4;                 // K-half selector
    const int idx   = lane & 15;                 // row / col index in tile

    v8f acc[4] = {};                             // 4 x (16x16 f32) accumulators

    for (int kk = 0; kk < F; kk += 4) {
        // A fragment: A[idx][kk + 2*half + {0,1}]  (ds_load_2addr_b64 pair)
        v2f a = *(const v2f*)&lds[(strip * 16 + idx) * LDSW + kk + 2 * half];
        #pragma unroll
        for (int j = 0; j < 4; ++j) {
            int col = (ch * 4 + j) * 16 + idx;
            // B fragment: B[kk + 2*half + {0,1}][col] (W is hot in L0/L2)
            v2f b;
            b.x = W[(size_t)(kk + 2 * half + 0) * F + col];
            b.y = W[(size_t)(kk + 2 * half + 1) * F + col];
            acc[j] = __builtin_amdgcn_wmma_f32_16x16x4_f32(
                false, a, false, b, (short)0, acc[j], false, false);
        }
    }

    // D layout: VGPR g -> row (strip*16 + g + 8*half), col = idx.
    // Fast path: one base pointer per tile, stores at immediate offsets g*512B.
    const int rowloc = strip * 16 + 8 * half;    // lane's first output row in tile
    if (full) {
        #pragma unroll
        for (int j = 0; j < 4; ++j) {
            int col = (ch * 4 + j) * 16 + idx;
            float* p = Y + (size_t)(rowbase + rowloc) * F + col;
            #pragma unroll
            for (int g8 = 0; g8 < 8; ++g8)
                p[g8 * F] = acc[j][g8];          // immediate offset stores
        }
    } else {
        #pragma unroll
        for (int j = 0; j < 4; ++j) {
            int col = (ch * 4 + j) * 16 + idx;
            #pragma unroll
            for (int g8 = 0; g8 < 8; ++g8) {
                int row = rowbase + rowloc + g8;
                if (row < N) Y[(size_t)row * F + col] = acc[j][g8];
            }
        }
    }
}

// ---------------------------------------------------------------------------
// Aggregation init: out[n][:] = bias + h[n][:] * dinv[n]^2   (self-loop term)
// ---------------------------------------------------------------------------
__global__ void k_agg_init(const float* __restrict__ Hin, const float* __restrict__ dinv,
                           const float* __restrict__ bias, float* __restrict__ out, int N) {
    int i = blockIdx.x * blockDim.x + threadIdx.x;   // float4 index
    if (i >= N * 32) return;
    int n = i >> 5;
    int c = (i & 31) * 4;
    float dv = dinv[n];
    float w = dv * dv;
    float4 h = *(const float4*)(Hin + (size_t)n * F + c);
    float4 b = *(const float4*)(bias + c);
    float4 o = make_float4(b.x + h.x * w, b.y + h.y * w, b.z + h.z * w, b.w + h.w * w);
    *(float4*)(out + (size_t)n * F + c) = o;
}

// ---------------------------------------------------------------------------
// Edge scatter: one wave per edge; lane handles 4 channels.
// out[dst][:] += h[src][:] * dinv[src]*dinv[dst]
// ---------------------------------------------------------------------------
__global__ void k_edge(const int* __restrict__ src, const int* __restrict__ dst,
                       const float* __restrict__ dinv, const float* __restrict__ h,
                       float* __restrict__ out, int E) {
    int wid  = (blockIdx.x * blockDim.x + threadIdx.x) >> 5;
    int lane = threadIdx.x & 31;
    int nw   = (gridDim.x * blockDim.x) >> 5;
    for (int e = wid; e < E; e += nw) {
        int s = src[e], d = dst[e];
        float w = dinv[s] * dinv[d];
        float4 v = *(const float4*)(h + (size_t)s * F + lane * 4);
        float* o = out + (size_t)d * F + lane * 4;
        atomicAdd(o + 0, v.x * w);
        atomicAdd(o + 1, v.y * w);
        atomicAdd(o + 2, v.z * w);
        atomicAdd(o + 3, v.w * w);
    }
}

// ---------------------------------------------------------------------------
// Classifier: wave per node; BN+ReLU fused; 2-class dot via shfl reduction.
// ---------------------------------------------------------------------------
__global__ void k_cls(const float* __restrict__ Hbuf, const float* __restrict__ Wc,
                      const float* __restrict__ bc,
                      const float* __restrict__ gg, const float* __restrict__ be,
                      const float* __restrict__ mm, const float* __restrict__ vv,
                      float* __restrict__ out, int N) {
    int wid  = (blockIdx.x * blockDim.x + threadIdx.x) >> 5;
    int lane = threadIdx.x & 31;
    int nw   = (gridDim.x * blockDim.x) >> 5;
    int c = lane * 4;
    float4 G = *(const float4*)(gg + c);
    float4 B = *(const float4*)(be + c);
    float4 M = *(const float4*)(mm + c);
    float4 V = *(const float4*)(vv + c);
    float s0 = G.x * rsqrtf(V.x + EPS_F), s1 = G.y * rsqrtf(V.y + EPS_F);
    float s2 = G.z * rsqrtf(V.z + EPS_F), s3 = G.w * rsqrtf(V.w + EPS_F);
    float w00 = Wc[(c + 0) * 2 + 0], w01 = Wc[(c + 0) * 2 + 1];
    float w10 = Wc[(c + 1) * 2 + 0], w11 = Wc[(c + 1) * 2 + 1];
    float w20 = Wc[(c + 2) * 2 + 0], w21 = Wc[(c + 2) * 2 + 1];
    float w30 = Wc[(c + 3) * 2 + 0], w31 = Wc[(c + 3) * 2 + 1];

    for (int n = wid; n < N; n += nw) {
        float4 h = *(const float4*)(Hbuf + (size_t)n * F + c);
        float h0 = fmaxf(0.f, (h.x - M.x) * s0 + B.x);
        float h1 = fmaxf(0.f, (h.y - M.y) * s1 + B.y);
        float h2 = fmaxf(0.f, (h.z - M.z) * s2 + B.z);
        float h3 = fmaxf(0.f, (h.w - M.w) * s3 + B.w);
        float p0 = h0 * w00 + h1 * w10 + h2 * w20 + h3 * w30;
        float p1 = h0 * w01 + h1 * w11 + h2 * w21 + h3 * w31;
        #pragma unroll
        for (int off = 16; off > 0; off >>= 1) {
            p0 += __shfl_down(p0, off);
            p1 += __shfl_down(p1, off);
        }
        if (lane == 0) {
            out[(size_t)n * 2 + 0] = p0 + bc[0];
            out[(size_t)n * 2 + 1] = p1 + bc[1];
        }
    }
}

// ---------------------------------------------------------------------------
// Launch
// ---------------------------------------------------------------------------
extern "C" void kernel_launch(void* const* d_in, const int* in_sizes, int n_in,
                              void* d_out, int out_size, void* d_ws, size_t ws_size,
                              hipStream_t stream) {
    const float* x   = (const float*)d_in[0];
    const int*   ei  = (const int*)d_in[1];
    const float* W1  = (const float*)d_in[2];
    const float* b1  = (const float*)d_in[3];
    const float* g1  = (const float*)d_in[4];
    const float* be1 = (const float*)d_in[5];
    const float* m1  = (const float*)d_in[6];
    const float* v1  = (const float*)d_in[7];
    const float* W2  = (const float*)d_in[8];
    const float* b2  = (const float*)d_in[9];
    const float* g2  = (const float*)d_in[10];
    const float* be2 = (const float*)d_in[11];
    const float* m2  = (const float*)d_in[12];
    const float* v2  = (const float*)d_in[13];
    const float* Wc  = (const float*)d_in[14];
    const float* bc  = (const float*)d_in[15];
    float* out = (float*)d_out;

    const int N = in_sizes[0] / F;
    const int E = in_sizes[1] / 2;
    const int* src = ei;        // edge_index[0]
    const int* dst = ei + E;    // edge_index[1]

    // Workspace layout: dinv[N] | bufA[N*128] | bufB[N*128]  (~52 MB)
    float* dinv = (float*)d_ws;
    size_t nAl  = ((size_t)N + 63) & ~(size_t)63;
    float* bufA = dinv + nAl;
    float* bufB = bufA + (size_t)N * F;

    const int T = 256;
    int gN  = (N + T - 1) / T;
    int gE  = (E + T - 1) / T;
    int gG  = (N + 63) / 64;            // GEMM: 64 rows / block
    int gA  = (N * 32 + T - 1) / T;     // agg init: float4 granularity
    int gEd = (E + 7) / 8;              // edge scatter: 8 waves / block
    int gC  = (N + 7) / 8;              // classifier: 8 waves / block

    // ---- normalization coefficients ----
    k_deg_init<<<gN, T, 0, stream>>>(dinv, N);
    k_deg_acc <<<gE, T, 0, stream>>>(dst, dinv, E);
    k_dinv    <<<gN, T, 0, stream>>>(dinv, N);

    // ---- layer 1 ----
    k_gemm<false><<<gG, T, 0, stream>>>(x, W1, bufA, N, nullptr, nullptr, nullptr, nullptr);
    k_agg_init<<<gA, T, 0, stream>>>(bufA, dinv, b1, bufB, N);
    k_edge    <<<gEd, T, 0, stream>>>(src, dst, dinv, bufA, bufB, E);

    // ---- layer 2 (BN1+ReLU fused into GEMM staging load) ----
    k_gemm<true><<<gG, T, 0, stream>>>(bufB, W2, bufA, N, g1, be1, m1, v1);
    k_agg_init<<<gA, T, 0, stream>>>(bufA, dinv, b2, bufB, N);
    k_edge    <<<gEd, T, 0, stream>>>(src, dst, dinv, bufA, bufB, E);

    // ---- classifier (BN2+ReLU fused) ----
    k_cls<<<gC, T, 0, stream>>>(bufB, Wc, bc, g2, be2, m2, v2, out, N);
}